// Blender_30786325578133
// MI455X (gfx1250) — compile-verified
//
#include <hip/hip_runtime.h>

// ---------------------------------------------------------------------------
// MI455X (gfx1250) fused style-blender.
// Memory-bound workload (~183 GFLOP fp32 vs ~0.95 GB HBM traffic @ 23.3 TB/s
// => ~40us floor). Keep fp32 precision and use V_WMMA_F32_16X16X4_F32 for all
// GEMMs; fuse the heavy fc MLP (656->512->512 per style layer) into a single
// one-pass-over-global_styles kernel with LDS-staged tiles (320KB/WGP).
// ---------------------------------------------------------------------------

typedef __attribute__((ext_vector_type(2))) float v2f;
typedef __attribute__((ext_vector_type(8))) float v8f;

#define S_    18
#define D_    512
#define BN_   32
#define GH_   128
#define AH_   16
#define FCH_  512
#define XW_   (GH_ + AH_ + D_)   // 656
#define GFAW_ (GH_ + AH_)        // 144

// wave32 fp32 WMMA: D(16x16 f32, v8f) = A(16x4 f32, v2f) * B(4x16 f32, v2f) + C
__device__ __forceinline__ v8f wmma4(v2f a, v2f b, v8f c) {
  return __builtin_amdgcn_wmma_f32_16x16x4_f32(
      /*neg_a=*/false, a, /*neg_b=*/false, b,
      /*c_mod=*/(short)0, c, /*reuse_a=*/false, /*reuse_b=*/false);
}

// ---------------------------------------------------------------------------
// Kernel 1: age MLP (1 -> 16 -> 16), trivial VALU work, writes af slice of gfa
// ---------------------------------------------------------------------------
__global__ void age_kernel(const float* __restrict__ ages,
                           const float* __restrict__ w1, const float* __restrict__ b1,
                           const float* __restrict__ w2, const float* __restrict__ b2,
                           float* __restrict__ gfa, int B) {
  int b = blockIdx.x * blockDim.x + threadIdx.x;
  if (b >= B) return;
  float a = ages[b];
  float h[AH_];
#pragma unroll
  for (int j = 0; j < AH_; ++j) h[j] = fmaxf(a * w1[j] + b1[j], 0.0f);
#pragma unroll
  for (int i = 0; i < AH_; ++i) {
    float acc = b2[i];
#pragma unroll
    for (int j = 0; j < AH_; ++j) acc += h[j] * w2[j * AH_ + i];
    gfa[(size_t)b * GFAW_ + GH_ + i] = acc;
  }
}

// ---------------------------------------------------------------------------
// Kernel 2: bn path per style layer s:
//   h = relu(gs[:,s,:] @ bn_w1[s] + b1) @ bn_w2[s] + b2   -> hout (B,S,32)
// grid = (B/32, S), block = 128 (4 waves). Wave w: (msub=w>>1, nsub=w&1).
// ---------------------------------------------------------------------------
__global__ void bn_kernel(const float* __restrict__ gs,
                          const float* __restrict__ w1, const float* __restrict__ b1,
                          const float* __restrict__ w2, const float* __restrict__ b2,
                          float* __restrict__ hout) {
  __shared__ float h1s[32][BN_ + 1];   // 32x32 relu'd intermediate (+pad)

  const int s     = blockIdx.y;
  const int mbase = blockIdx.x * 32;
  const int wave  = threadIdx.x >> 5;
  const int lane  = threadIdx.x & 31;
  const int lrow  = lane & 15;
  const int lhi   = lane >> 4;         // 0: K 0..1, 1: K 2..3
  const int msub  = wave >> 1;
  const int nsub  = wave & 1;
  const int m0    = mbase + 16 * msub;
  const int n0    = 16 * nsub;

  const float* gsrow = gs + (size_t)(m0 + lrow) * (S_ * D_) + (size_t)s * D_;
  const float* w1s   = w1 + (size_t)s * D_ * BN_;

  // GEMM1: [16,512] x [512,16] tile, K-step 4
  v8f acc = {};
  for (int k = 0; k < D_; k += 4) {
    const int ka = k + 2 * lhi;
    v2f a;  a.x = gsrow[ka];                      a.y = gsrow[ka + 1];
    v2f bb; bb.x = w1s[(size_t)ka * BN_ + n0 + lrow];
            bb.y = w1s[(size_t)(ka + 1) * BN_ + n0 + lrow];
    acc = wmma4(a, bb, acc);
  }
  const float* b1s = b1 + s * BN_;
#pragma unroll
  for (int i = 0; i < 8; ++i) {
    const int m = 16 * msub + i + 8 * lhi;
    const int n = n0 + lrow;
    h1s[m][n] = fmaxf(acc[i] + b1s[n], 0.0f);
  }
  __syncthreads();

  // GEMM2: [16,32] x [32,16] tile
  const float* w2s = w2 + (size_t)s * BN_ * BN_;
  v8f acc2 = {};
#pragma unroll
  for (int k = 0; k < BN_; k += 4) {
    const int ka = k + 2 * lhi;
    v2f a;  a.x = h1s[16 * msub + lrow][ka];      a.y = h1s[16 * msub + lrow][ka + 1];
    v2f bb; bb.x = w2s[(size_t)ka * BN_ + n0 + lrow];
            bb.y = w2s[(size_t)(ka + 1) * BN_ + n0 + lrow];
    acc2 = wmma4(a, bb, acc2);
  }
  const float* b2s = b2 + s * BN_;
#pragma unroll
  for (int i = 0; i < 8; ++i) {
    const int b = m0 + i + 8 * lhi;
    const int n = n0 + lrow;
    hout[(size_t)b * (S_ * BN_) + (size_t)s * BN_ + n] = acc2[i] + b2s[n];
  }
}

// ---------------------------------------------------------------------------
// Kernel 3: gm path: gf = relu(h.reshape(B,576) @ gm_w1 + b1) @ gm_w2 + b2
// grid = B/32, block = 128 (4 waves). Wave owns a 32-col slab of the 128 cols.
// ---------------------------------------------------------------------------
__global__ void gm_kernel(const float* __restrict__ hflat,
                          const float* __restrict__ w1, const float* __restrict__ b1,
                          const float* __restrict__ w2, const float* __restrict__ b2,
                          float* __restrict__ gfa) {
  __shared__ float g1s[32][GH_ + 4];   // 32x128 relu'd intermediate (+pad)

  const int mbase = blockIdx.x * 32;
  const int wave  = threadIdx.x >> 5;
  const int lane  = threadIdx.x & 31;
  const int lrow  = lane & 15;
  const int lhi   = lane >> 4;
  const int nb    = wave * 32;
  const int K1    = S_ * BN_;          // 576

  const float* r0 = hflat + (size_t)(mbase + lrow) * K1;
  const float* r1 = hflat + (size_t)(mbase + 16 + lrow) * K1;

  v8f acc[2][2] = {};
  for (int k = 0; k < K1; k += 4) {
    const int ka = k + 2 * lhi;
    v2f a0; a0.x = r0[ka]; a0.y = r0[ka + 1];
    v2f a1; a1.x = r1[ka]; a1.y = r1[ka + 1];
    v2f bv[2];
#pragma unroll
    for (int j = 0; j < 2; ++j) {
      const int n = nb + 16 * j + lrow;
      bv[j].x = w1[(size_t)ka * GH_ + n];
      bv[j].y = w1[(size_t)(ka + 1) * GH_ + n];
    }
#pragma unroll
    for (int j = 0; j < 2; ++j) {
      acc[0][j] = wmma4(a0, bv[j], acc[0][j]);
      acc[1][j] = wmma4(a1, bv[j], acc[1][j]);
    }
  }
#pragma unroll
  for (int ms = 0; ms < 2; ++ms)
#pragma unroll
    for (int ns = 0; ns < 2; ++ns)
#pragma unroll
      for (int i = 0; i < 8; ++i) {
        const int m = 16 * ms + i + 8 * lhi;
        const int n = nb + 16 * ns + lrow;
        g1s[m][n] = fmaxf(acc[ms][ns][i] + b1[n], 0.0f);
      }
  __syncthreads();

  v8f acc2[2][2] = {};
#pragma unroll 4
  for (int k = 0; k < GH_; k += 4) {
    const int ka = k + 2 * lhi;
    v2f a0; a0.x = g1s[lrow][ka];      a0.y = g1s[lrow][ka + 1];
    v2f a1; a1.x = g1s[16 + lrow][ka]; a1.y = g1s[16 + lrow][ka + 1];
    v2f bv[2];
#pragma unroll
    for (int j = 0; j < 2; ++j) {
      const int n = nb + 16 * j + lrow;
      bv[j].x = w2[(size_t)ka * GH_ + n];
      bv[j].y = w2[(size_t)(ka + 1) * GH_ + n];
    }
#pragma unroll
    for (int j = 0; j < 2; ++j) {
      acc2[0][j] = wmma4(a0, bv[j], acc2[0][j]);
      acc2[1][j] = wmma4(a1, bv[j], acc2[1][j]);
    }
  }
#pragma unroll
  for (int ms = 0; ms < 2; ++ms)
#pragma unroll
    for (int ns = 0; ns < 2; ++ns)
#pragma unroll
      for (int i = 0; i < 8; ++i) {
        const int b = mbase + 16 * ms + i + 8 * lhi;
        const int n = nb + 16 * ns + lrow;
        gfa[(size_t)b * GFAW_ + n] = acc2[ms][ns][i] + b2[n];
      }
}

// ---------------------------------------------------------------------------
// Kernel 4: fc path (dominant): per (s, 32-row tile):
//   X = [gfa | gs]  (32 x 656, staged in LDS once — single HBM pass over gs)
//   Y = relu(X @ fc_w1[s] + b1)      (32 x 512, staged in LDS)
//   out = Y @ fc_w2[s] + b2 + gs     (residual read back from the X tile)
// block = 256 (8 waves), wave owns a 64-col slab: 2(M) x 4(N) v8f accumulators.
// LDS strides padded (660 / 516) for conflict-free A-fragment reads.
// ---------------------------------------------------------------------------
#define XP_ 660
#define YP_ 516
#define FC_LDS_BYTES ((32 * XP_ + 32 * YP_) * 4)   // 150,528 B < 320 KB/WGP

__global__ void fc_kernel(const float* __restrict__ gs,
                          const float* __restrict__ gfa,
                          const float* __restrict__ w1, const float* __restrict__ b1,
                          const float* __restrict__ w2, const float* __restrict__ b2,
                          float* __restrict__ out) {
  extern __shared__ float smem[];
  float* xs = smem;              // [32][XP_]
  float* ys = smem + 32 * XP_;   // [32][YP_]

  const int s     = blockIdx.y;
  const int mbase = blockIdx.x * 32;
  const int tid   = threadIdx.x;
  const int wave  = tid >> 5;
  const int lane  = tid & 31;
  const int lrow  = lane & 15;
  const int lhi   = lane >> 4;
  const int nb    = wave * 64;   // wave's 64-col slab of the 512 outputs

  // ---- stage X tile: gfa part (32 x 144) ----
  for (int idx = tid; idx < 32 * GFAW_; idx += 256) {
    const int r = idx / GFAW_, c = idx % GFAW_;
    xs[r * XP_ + c] = gfa[(size_t)(mbase + r) * GFAW_ + c];
  }
  // ---- stage X tile: global_styles part (32 x 512) as float4 ----
  for (int idx = tid; idx < 32 * (D_ / 4); idx += 256) {
    const int r = idx / (D_ / 4), c4 = idx % (D_ / 4);
    const float4 v =
        ((const float4*)(gs + (size_t)(mbase + r) * (S_ * D_) + (size_t)s * D_))[c4];
    float* dst = xs + r * XP_ + GFAW_ + c4 * 4;
    dst[0] = v.x; dst[1] = v.y; dst[2] = v.z; dst[3] = v.w;
  }
  __syncthreads();

  // ---- GEMM1: [32,656] x [656,64-slab], K-step 4 ----
  const float* w1s = w1 + (size_t)s * XW_ * FCH_;
  v8f acc[2][4] = {};
  for (int k = 0; k < XW_; k += 4) {
    const int ka = k + 2 * lhi;
    v2f a0; a0.x = xs[lrow * XP_ + ka];        a0.y = xs[lrow * XP_ + ka + 1];
    v2f a1; a1.x = xs[(16 + lrow) * XP_ + ka]; a1.y = xs[(16 + lrow) * XP_ + ka + 1];
    v2f bv[4];
#pragma unroll
    for (int j = 0; j < 4; ++j) {
      const int n = nb + 16 * j + lrow;
      bv[j].x = w1s[(size_t)ka * FCH_ + n];
      bv[j].y = w1s[(size_t)(ka + 1) * FCH_ + n];
    }
#pragma unroll
    for (int j = 0; j < 4; ++j) {
      acc[0][j] = wmma4(a0, bv[j], acc[0][j]);
      acc[1][j] = wmma4(a1, bv[j], acc[1][j]);
    }
  }
  // bias + relu -> LDS Y tile
  const float* b1s = b1 + s * FCH_;
#pragma unroll
  for (int ms = 0; ms < 2; ++ms)
#pragma unroll
    for (int j = 0; j < 4; ++j)
#pragma unroll
      for (int i = 0; i < 8; ++i) {
        const int m = 16 * ms + i + 8 * lhi;
        const int n = nb + 16 * j + lrow;
        ys[m * YP_ + n] = fmaxf(acc[ms][j][i] + b1s[n], 0.0f);
      }
  __syncthreads();

  // ---- GEMM2: [32,512] x [512,64-slab] ----
  const float* w2s = w2 + (size_t)s * FCH_ * D_;
#pragma unroll
  for (int ms = 0; ms < 2; ++ms)
#pragma unroll
    for (int j = 0; j < 4; ++j) acc[ms][j] = (v8f){};
  for (int k = 0; k < FCH_; k += 4) {
    const int ka = k + 2 * lhi;
    v2f a0; a0.x = ys[lrow * YP_ + ka];        a0.y = ys[lrow * YP_ + ka + 1];
    v2f a1; a1.x = ys[(16 + lrow) * YP_ + ka]; a1.y = ys[(16 + lrow) * YP_ + ka + 1];
    v2f bv[4];
#pragma unroll
    for (int j = 0; j < 4; ++j) {
      const int n = nb + 16 * j + lrow;
      bv[j].x = w2s[(size_t)ka * D_ + n];
      bv[j].y = w2s[(size_t)(ka + 1) * D_ + n];
    }
#pragma unroll
    for (int j = 0; j < 4; ++j) {
      acc[0][j] = wmma4(a0, bv[j], acc[0][j]);
      acc[1][j] = wmma4(a1, bv[j], acc[1][j]);
    }
  }
  // bias + residual (gs slice lives in the staged X tile) -> out
  const float* b2s = b2 + s * D_;
#pragma unroll
  for (int ms = 0; ms < 2; ++ms)
#pragma unroll
    for (int j = 0; j < 4; ++j)
#pragma unroll
      for (int i = 0; i < 8; ++i) {
        const int m = 16 * ms + i + 8 * lhi;
        const int n = nb + 16 * j + lrow;
        const float res = acc[ms][j][i] + b2s[n] + xs[m * XP_ + GFAW_ + n];
        out[(size_t)(mbase + m) * (S_ * D_) + (size_t)s * D_ + n] = res;
      }
}

// ---------------------------------------------------------------------------
// Host-side launch. Input order (from setup_inputs):
//  0 local_styles (unused)  1 global_styles  2 target_ages
//  3 bn_w1  4 bn_b1  5 bn_w2  6 bn_b2
//  7 gm_w1  8 gm_b1  9 gm_w2 10 gm_b2
// 11 age_w1 12 age_b1 13 age_w2 14 age_b2
// 15 fc_w1 16 fc_b1 17 fc_w2 18 fc_b2
// ---------------------------------------------------------------------------
extern "C" void kernel_launch(void* const* d_in, const int* in_sizes, int n_in,
                              void* d_out, int out_size, void* d_ws, size_t ws_size,
                              hipStream_t stream) {
  const float* gs     = (const float*)d_in[1];
  const float* ages   = (const float*)d_in[2];
  const float* bn_w1  = (const float*)d_in[3];
  const float* bn_b1  = (const float*)d_in[4];
  const float* bn_w2  = (const float*)d_in[5];
  const float* bn_b2  = (const float*)d_in[6];
  const float* gm_w1  = (const float*)d_in[7];
  const float* gm_b1  = (const float*)d_in[8];
  const float* gm_w2  = (const float*)d_in[9];
  const float* gm_b2  = (const float*)d_in[10];
  const float* age_w1 = (const float*)d_in[11];
  const float* age_b1 = (const float*)d_in[12];
  const float* age_w2 = (const float*)d_in[13];
  const float* age_b2 = (const float*)d_in[14];
  const float* fc_w1  = (const float*)d_in[15];
  const float* fc_b1  = (const float*)d_in[16];
  const float* fc_w2  = (const float*)d_in[17];
  const float* fc_b2  = (const float*)d_in[18];
  float* outp = (float*)d_out;

  const int B = in_sizes[2];          // 8192

  float* gfa = (float*)d_ws;                    // (B, 144)
  float* h   = gfa + (size_t)B * GFAW_;         // (B, S, 32) == (B, 576)

  age_kernel<<<(B + 255) / 256, 256, 0, stream>>>(ages, age_w1, age_b1,
                                                  age_w2, age_b2, gfa, B);
  bn_kernel<<<dim3(B / 32, S_), 128, 0, stream>>>(gs, bn_w1, bn_b1,
                                                  bn_w2, bn_b2, h);
  gm_kernel<<<B / 32, 128, 0, stream>>>(h, gm_w1, gm_b1, gm_w2, gm_b2, gfa);
  fc_kernel<<<dim3(B / 32, S_), 256, FC_LDS_BYTES, stream>>>(
      gs, gfa, fc_w1, fc_b1, fc_w2, fc_b2, outp);
}